// Indexer_80659485819390
// MI455X (gfx1250) — compile-verified
//
#include <hip/hip_runtime.h>
#include <math.h>

typedef __attribute__((ext_vector_type(2))) float v2f;
typedef __attribute__((ext_vector_type(4))) float v4f;
typedef __attribute__((ext_vector_type(8))) float v8f;

constexpr int B_ = 16, T_ = 64, M_ = 2048, D_ = 64, C_ = 256;
constexpr int NC = 70;              // ctrl columns (64+1+1+3+1)
constexpr int NBT = B_ * T_;        // 1024
constexpr float EPSV = 1e-8f;

// ---------------------------------------------------------------------------
// Kernel 1: ctrl = control_state @ W + b   via v_wmma_f32_16x16x4_f32
// Grid: 64 row-tiles x 5 col-tiles, one wave (32 threads) per 16x16 tile.
// ---------------------------------------------------------------------------
__global__ void proj_wmma(const float* __restrict__ CS,   // [1024,256]
                          const float* __restrict__ W,    // [256,70]
                          const float* __restrict__ bias, // [70]
                          float* __restrict__ ctrl) {     // [1024,70]
  const int tile = blockIdx.x;
  const int rowBase = (tile / 5) * 16;
  const int colBase = (tile % 5) * 16;
  const int lane = threadIdx.x;
  const int half = lane >> 4;   // selects K offset {0,2}
  const int sub  = lane & 15;   // M (for A) / N (for B)
  const int col  = colBase + sub;
  const int colC = col < NC ? col : (NC - 1);
  const float colMask = col < NC ? 1.0f : 0.0f;

  const float* arow = CS + (size_t)(rowBase + sub) * C_;
  v8f acc = {};
  for (int k0 = 0; k0 < C_; k0 += 4) {
    const int ka = k0 + 2 * half;
    v2f a, b;
    a.x = arow[ka];
    a.y = arow[ka + 1];
    b.x = W[(size_t)ka * NC + colC] * colMask;        // branch-free: EXEC stays full
    b.y = W[(size_t)(ka + 1) * NC + colC] * colMask;
    acc = __builtin_amdgcn_wmma_f32_16x16x4_f32(
        /*neg_a=*/false, a, /*neg_b=*/false, b,
        /*c_mod=*/(short)0, acc, /*reuse_a=*/false, /*reuse_b=*/false);
  }
  if (col < NC) {
    const float bv = bias[col];
#pragma unroll
    for (int v = 0; v < 8; ++v) {
      const int row = rowBase + v + 8 * half;   // D layout: VGPR v -> M = v (+8 for hi half)
      ctrl[(size_t)row * NC + col] = acc[v] + bv;
    }
  }
}

// ---------------------------------------------------------------------------
// Kernel 2: per-(b,t) activations -> params[bt*8] = {strength/knorm, interp,
//           s0, s1, s2, p=2+softplus, _, _}
// ---------------------------------------------------------------------------
__device__ __forceinline__ float softplusf(float x) {
  return fmaxf(x, 0.0f) + log1pf(expf(-fabsf(x)));   // = logaddexp(x, 0)
}

__global__ void params_kernel(const float* __restrict__ ctrl,
                              float* __restrict__ params) {
  const int bt = blockIdx.x * blockDim.x + threadIdx.x;
  if (bt >= NBT) return;
  const float* c = ctrl + (size_t)bt * NC;
  float ss = 0.0f;
#pragma unroll
  for (int d = 0; d < D_; ++d) { const float v = c[d]; ss += v * v; }
  const float knorm = fmaxf(sqrtf(ss), EPSV);
  const float strength = softplusf(c[64]);
  const float interp = 1.0f / (1.0f + expf(-c[65]));
  const float s0 = c[66], s1 = c[67], s2 = c[68];
  const float mx = fmaxf(s0, fmaxf(s1, s2));
  const float e0 = expf(s0 - mx), e1 = expf(s1 - mx), e2 = expf(s2 - mx);
  const float inv = 1.0f / (e0 + e1 + e2);
  float* o = params + (size_t)bt * 8;
  o[0] = strength / knorm;
  o[1] = interp;
  o[2] = e0 * inv;
  o[3] = e1 * inv;
  o[4] = e2 * inv;
  o[5] = 2.0f + softplusf(c[69]);
}

// ---------------------------------------------------------------------------
// Kernel 3 (HBM-bound: streams 536 MB of `memory` exactly once).
// z[bt,m] = strength*dot/(knorm*mnorm).
// One block per (b,t); each wave loads 512B contiguous per request (two
// 64-float rows), NON-TEMPORAL so the single-use stream doesn't thrash the
// 192MB L2. 2x unrolled: two independent load->reduce chains in flight.
// ---------------------------------------------------------------------------
__global__ void content_kernel(const float* __restrict__ ctrl,
                               const float* __restrict__ params,
                               const float* __restrict__ memory,
                               float* __restrict__ z) {
  const int bt = blockIdx.x;
  __shared__ __align__(16) float key[D_];
  const int tid = threadIdx.x;
  if (tid < D_) key[tid] = ctrl[(size_t)bt * NC + tid];
  __syncthreads();

  const float scale = params[(size_t)bt * 8 + 0];
  const int lane = tid & 31;
  const int wave = tid >> 5;        // 0..7
  const int half = lane >> 4;       // row within the pair
  const int sub  = lane & 15;       // d/4
  const v4f k4 = *reinterpret_cast<const v4f*>(&key[sub * 4]);
  const float* memBase = memory + (size_t)bt * M_ * D_;
  float* zrow = z + (size_t)bt * M_;

  // rows per iteration: wave handles {m0+half, m0+16+half}; 8 waves cover 32
  // consecutive rows per step, loop covers 2048 rows in 64 steps.
  for (int m0 = wave * 2; m0 < M_; m0 += 32) {
    const int rowA = m0 + half;
    const int rowB = m0 + 16 + half;
    const v4f mvA = __builtin_nontemporal_load(
        reinterpret_cast<const v4f*>(memBase + (size_t)rowA * D_ + sub * 4));
    const v4f mvB = __builtin_nontemporal_load(
        reinterpret_cast<const v4f*>(memBase + (size_t)rowB * D_ + sub * 4));

    float dotA = mvA.x * k4.x + mvA.y * k4.y + mvA.z * k4.z + mvA.w * k4.w;
    float ssqA = mvA.x * mvA.x + mvA.y * mvA.y + mvA.z * mvA.z + mvA.w * mvA.w;
    float dotB = mvB.x * k4.x + mvB.y * k4.y + mvB.z * k4.z + mvB.w * k4.w;
    float ssqB = mvB.x * mvB.x + mvB.y * mvB.y + mvB.z * mvB.z + mvB.w * mvB.w;
#pragma unroll
    for (int off = 8; off >= 1; off >>= 1) {   // reduce within 16-lane halves
      dotA += __shfl_xor(dotA, off, 32);
      ssqA += __shfl_xor(ssqA, off, 32);
      dotB += __shfl_xor(dotB, off, 32);
      ssqB += __shfl_xor(ssqB, off, 32);
    }
    if (sub == 0) {
      zrow[rowA] = scale * dotA / fmaxf(sqrtf(ssqA), EPSV);
      zrow[rowB] = scale * dotB / fmaxf(sqrtf(ssqB), EPSV);
    }
  }
}

// ---------------------------------------------------------------------------
// Kernel 4: softmax over T (axis=-2 quirk) + interpolation, in-place on z.
// One thread per (b,m); 64 logits held in registers.
// ---------------------------------------------------------------------------
__global__ void tsoftmax_kernel(float* __restrict__ z,
                                const float* __restrict__ params,
                                const float* __restrict__ prior) {
  const int idx = blockIdx.x * blockDim.x + threadIdx.x;   // over B*M
  const int b = idx / M_;
  const int m = idx % M_;
  float* zp = z + (size_t)b * T_ * M_ + m;
  const float* pr = prior + (size_t)b * T_ * M_ + m;

  float vals[T_];
  float mx = -INFINITY;
#pragma unroll
  for (int t = 0; t < T_; ++t) {
    vals[t] = zp[(size_t)t * M_];
    mx = fmaxf(mx, vals[t]);
  }
  float sum = 0.0f;
#pragma unroll
  for (int t = 0; t < T_; ++t) {
    vals[t] = expf(vals[t] - mx);
    sum += vals[t];
  }
  const float inv = 1.0f / sum;
#pragma unroll
  for (int t = 0; t < T_; ++t) {
    const float g = params[(size_t)(b * T_ + t) * 8 + 1];
    zp[(size_t)t * M_] = g * (vals[t] * inv) + (1.0f - g) * pr[(size_t)t * M_];
  }
}

// ---------------------------------------------------------------------------
// Kernel 5: circular shift (roll -1,0,+1) + sharpening w^p + normalize over M.
// One block (256 thr) per (b,t); row staged in LDS, wrap via & (M-1).
// ---------------------------------------------------------------------------
__global__ void shift_kernel(const float* __restrict__ w,
                             const float* __restrict__ params,
                             float* __restrict__ out) {
  const int bt = blockIdx.x;
  __shared__ float lw[M_];
  __shared__ float red[8];
  const int tid = threadIdx.x;
  const float* wrow = w + (size_t)bt * M_;
  for (int m = tid; m < M_; m += 256) lw[m] = wrow[m];
  __syncthreads();

  const float s0 = params[(size_t)bt * 8 + 2];
  const float s1 = params[(size_t)bt * 8 + 3];
  const float s2 = params[(size_t)bt * 8 + 4];
  const float p  = params[(size_t)bt * 8 + 5];

  float vals[8];
  float lsum = 0.0f;
#pragma unroll
  for (int i = 0; i < 8; ++i) {
    const int m = tid + i * 256;
    // roll(w,-1)[m]=w[m+1] (weight s0), roll(w,0)=w[m] (s1), roll(w,1)[m]=w[m-1] (s2)
    const float wn = s0 * lw[(m + 1) & (M_ - 1)] + s1 * lw[m] +
                     s2 * lw[(m - 1) & (M_ - 1)];
    const float sv = powf(wn, p);
    vals[i] = sv;
    lsum += sv;
  }
#pragma unroll
  for (int off = 16; off >= 1; off >>= 1) lsum += __shfl_xor(lsum, off, 32);
  if ((tid & 31) == 0) red[tid >> 5] = lsum;
  __syncthreads();
  float tot = 0.0f;
#pragma unroll
  for (int i = 0; i < 8; ++i) tot += red[i];
  const float inv = 1.0f / tot;
  float* orow = out + (size_t)bt * M_;
#pragma unroll
  for (int i = 0; i < 8; ++i) orow[tid + i * 256] = vals[i] * inv;
}

// ---------------------------------------------------------------------------
extern "C" void kernel_launch(void* const* d_in, const int* in_sizes, int n_in,
                              void* d_out, int out_size, void* d_ws, size_t ws_size,
                              hipStream_t stream) {
  const float* CS     = (const float*)d_in[0];  // [B,T,C]
  const float* memory = (const float*)d_in[1];  // [B,T,M,D]
  const float* prior  = (const float*)d_in[2];  // [B,T,M]
  const float* W      = (const float*)d_in[3];  // [C,70]
  const float* bias   = (const float*)d_in[4];  // [70]
  float* out = (float*)d_out;

  float* ctrl   = (float*)d_ws;                 // 1024*70 floats
  float* params = ctrl + (size_t)NBT * NC;      // 1024*8 floats
  float* z      = params + (size_t)NBT * 8;     // B*T*M floats (reused in-place as w)

  proj_wmma<<<(NBT / 16) * 5, 32, 0, stream>>>(CS, W, bias, ctrl);
  params_kernel<<<NBT / 256, 256, 0, stream>>>(ctrl, params);
  content_kernel<<<NBT, 256, 0, stream>>>(ctrl, params, memory, z);
  tsoftmax_kernel<<<(B_ * M_) / 256, 256, 0, stream>>>(z, params, prior);
  shift_kernel<<<NBT, 256, 0, stream>>>(z, params, out);
}